// GraphSAGE_Net_39238821216833
// MI455X (gfx1250) — compile-verified
//
#include <hip/hip_runtime.h>

#define N_NODES 50000
#define F_IN    128
#define H_DIM   256
#define C_DIM   16
#define M_TILES (N_NODES / 16)   // 3125

typedef float v2f __attribute__((ext_vector_type(2)));
typedef float v8f __attribute__((ext_vector_type(8)));

// ---------------------------------------------------------------------------
// Zero the scratch regions (deg + agg1 + agg2 are laid out contiguously).
// ---------------------------------------------------------------------------
__global__ void zero_ws_kernel(float4* __restrict__ p, int n4) {
    int i = blockIdx.x * blockDim.x + threadIdx.x;
    if (i < n4) {
        float4 z; z.x = 0.f; z.y = 0.f; z.z = 0.f; z.w = 0.f;
        p[i] = z;
    }
}

// ---------------------------------------------------------------------------
// Layer-1 message scatter: agg1[dst] += x[src], deg[dst] += 1.
// One thread per (edge, 4-feature group); float4 load (coalesced: 32 lanes
// cover one contiguous 512B row), 4 fp32 atomics (agg1 is L2-resident).
// ---------------------------------------------------------------------------
__global__ void scatter_f128_kernel(const float* __restrict__ x,
                                    const int* __restrict__ src,
                                    const int* __restrict__ dst,
                                    float* __restrict__ agg,
                                    float* __restrict__ deg,
                                    int E) {
    int tid = blockIdx.x * blockDim.x + threadIdx.x;
    if (tid >= E * 32) return;          // E*32 = 19.2M < 2^31
    int e = tid >> 5;
    int g = tid & 31;                   // 32 groups of 4 floats = 128 feats
    int s = src[e];
    int d = dst[e];
    if (g == 0) atomicAdd(&deg[d], 1.0f);
    const float4 v = *(const float4*)(x + (long long)s * F_IN + g * 4);
    float* o = agg + (long long)d * F_IN + g * 4;
    atomicAdd(o + 0, v.x);
    atomicAdd(o + 1, v.y);
    atomicAdd(o + 2, v.z);
    atomicAdd(o + 3, v.w);
}

// ---------------------------------------------------------------------------
// Layer-2 message scatter: agg2[dst] += h[src] (256 features).
// ---------------------------------------------------------------------------
__global__ void scatter_f256_kernel(const float* __restrict__ h,
                                    const int* __restrict__ src,
                                    const int* __restrict__ dst,
                                    float* __restrict__ agg,
                                    int E) {
    int tid = blockIdx.x * blockDim.x + threadIdx.x;
    if (tid >= E * 64) return;          // E*64 = 38.4M < 2^31
    int e = tid >> 6;
    int g = tid & 63;                   // 64 groups of 4 floats = 256 feats
    int s = src[e];
    int d = dst[e];
    const float4 v = *(const float4*)(h + (long long)s * H_DIM + g * 4);
    float* o = agg + (long long)d * H_DIM + g * 4;
    atomicAdd(o + 0, v.x);
    atomicAdd(o + 1, v.y);
    atomicAdd(o + 2, v.z);
    atomicAdd(o + 3, v.w);
}

// ---------------------------------------------------------------------------
// Layer 1: h = relu((agg1/deg) @ W1_l + b1 + x @ W1_r)
//
// Blocked mapping:
//   blockIdx.x & 1       -> N-tile group (cols [grp*128, grp*128+128))
//   (blockIdx.x >> 1)*8  -> base M-tile; wave w handles tileM = base + w
// Each wave keeps 8 accumulators (16x128 output strip): one A fragment
// (global_load_b64) feeds 8 v_wmma_f32_16x16x4_f32, and all 8 waves of a
// block issue IDENTICAL B loads (same N group) -> WGP$/L0 hits after the
// first wave. 512 WMMAs per wave.
//
// fp32 WMMA operand layouts (ISA 7.12.2):
//   A 16x4 : lane&15 = M row; a.x/a.y = K = kb, kb+1, kb = 4*kk + 2*(lane>=16)
//   B 4x16 : lane&15 = N col; b.x/b.y = same K pair
//   C/D    : 8 VGPRs; elem v in lane l -> M = v + 8*(l>=16), N = l&15
// ---------------------------------------------------------------------------
__global__ __launch_bounds__(256) void sage_gemm1_kernel(
        const float* __restrict__ x,
        const float* __restrict__ agg1,
        const float* __restrict__ deg,
        const float* __restrict__ W_l,
        const float* __restrict__ W_r,
        const float* __restrict__ bias,
        float* __restrict__ h) {
    const int lane  = threadIdx.x & 31;
    const int wv    = threadIdx.x >> 5;            // 0..7
    const int grp   = blockIdx.x & 1;              // which 128-col half
    const int tileM = (blockIdx.x >> 1) * 8 + wv;
    if (tileM >= M_TILES) return;                  // uniform per wave

    const int r       = lane & 15;
    const int row     = tileM * 16 + r;            // A row this lane supplies
    const int colbase = grp * 128 + r;             // col of N-tile t = +16*t
    const int khalf   = (lane >> 4) << 1;          // 0 or 2

    const float invd = 1.0f / fmaxf(deg[row], 1.0f);
    const float* arow = agg1 + (long long)row * F_IN;
    const float* xrow = x    + (long long)row * F_IN;

    v8f acc[8];
#pragma unroll
    for (int t = 0; t < 8; ++t)
        acc[t] = (v8f){0.f, 0.f, 0.f, 0.f, 0.f, 0.f, 0.f, 0.f};

    // term 1: (agg1/deg) @ W_l
#pragma unroll 4
    for (int kk = 0; kk < F_IN / 4; ++kk) {
        const int kb = kk * 4 + khalf;
        v2f a;
        a.x = arow[kb]     * invd;
        a.y = arow[kb + 1] * invd;
        const float* w0 = W_l + (long long)kb * H_DIM + colbase;
#pragma unroll
        for (int t = 0; t < 8; ++t) {
            v2f b;
            b.x = w0[t * 16];
            b.y = w0[H_DIM + t * 16];
            acc[t] = __builtin_amdgcn_wmma_f32_16x16x4_f32(
                         false, a, false, b, (short)0, acc[t], false, false);
        }
    }
    // term 2: x @ W_r
#pragma unroll 4
    for (int kk = 0; kk < F_IN / 4; ++kk) {
        const int kb = kk * 4 + khalf;
        v2f a;
        a.x = xrow[kb];
        a.y = xrow[kb + 1];
        const float* w0 = W_r + (long long)kb * H_DIM + colbase;
#pragma unroll
        for (int t = 0; t < 8; ++t) {
            v2f b;
            b.x = w0[t * 16];
            b.y = w0[H_DIM + t * 16];
            acc[t] = __builtin_amdgcn_wmma_f32_16x16x4_f32(
                         false, a, false, b, (short)0, acc[t], false, false);
        }
    }

    const int hi = (lane >> 4) << 3;   // +8 rows for upper half-wave
#pragma unroll
    for (int t = 0; t < 8; ++t) {
        const int c  = colbase + t * 16;
        const float bc = bias[c];
#pragma unroll
        for (int v = 0; v < 8; ++v) {
            const int orow = tileM * 16 + v + hi;
            h[(long long)orow * H_DIM + c] = fmaxf(acc[t][v] + bc, 0.0f);
        }
    }
}

// ---------------------------------------------------------------------------
// Layer 2 + log_softmax: out = log_softmax((agg2/deg) @ W2_l + b2 + h @ W2_r)
// C=16 == one N tile, K=256 -> 128 WMMAs per wave. B panels (32 KB total)
// are identical for every wave -> L0-hot. Softmax reductions via shfl_xor
// across the 16-lane halves (masks < 16 never cross a half; wave32-correct).
// ---------------------------------------------------------------------------
__global__ __launch_bounds__(256) void sage_gemm2_softmax_kernel(
        const float* __restrict__ h,
        const float* __restrict__ agg2,
        const float* __restrict__ deg,
        const float* __restrict__ W_l,
        const float* __restrict__ W_r,
        const float* __restrict__ bias,
        float* __restrict__ out) {
    const int lane = threadIdx.x & 31;
    const int wave = (blockIdx.x * blockDim.x + threadIdx.x) >> 5;
    if (wave >= M_TILES) return;       // whole wave exits together
    const int tileM = wave;
    const int r     = lane & 15;
    const int row   = tileM * 16 + r;
    const int col   = r;               // C = 16 columns
    const int khalf = (lane >> 4) << 1;

    const float invd = 1.0f / fmaxf(deg[row], 1.0f);
    const float* arow = agg2 + (long long)row * H_DIM;
    const float* hrow = h    + (long long)row * H_DIM;

    v8f acc = {0.f, 0.f, 0.f, 0.f, 0.f, 0.f, 0.f, 0.f};

#pragma unroll 8
    for (int kk = 0; kk < H_DIM / 4; ++kk) {
        const int kb = kk * 4 + khalf;
        v2f a, b;
        a.x = arow[kb]     * invd;
        a.y = arow[kb + 1] * invd;
        b.x = W_l[kb * C_DIM + col];
        b.y = W_l[(kb + 1) * C_DIM + col];
        acc = __builtin_amdgcn_wmma_f32_16x16x4_f32(
                  false, a, false, b, (short)0, acc, false, false);
    }
#pragma unroll 8
    for (int kk = 0; kk < H_DIM / 4; ++kk) {
        const int kb = kk * 4 + khalf;
        v2f a, b;
        a.x = hrow[kb];
        a.y = hrow[kb + 1];
        b.x = W_r[kb * C_DIM + col];
        b.y = W_r[(kb + 1) * C_DIM + col];
        acc = __builtin_amdgcn_wmma_f32_16x16x4_f32(
                  false, a, false, b, (short)0, acc, false, false);
    }

    const float bcol = bias[col];
    const int hi = (lane >> 4) << 3;
#pragma unroll
    for (int v = 0; v < 8; ++v) {
        float z = acc[v] + bcol;
        float m = z;
        for (int d = 1; d < 16; d <<= 1)
            m = fmaxf(m, __shfl_xor(m, d, 32));
        float ex = __expf(z - m);
        float s = ex;
        for (int d = 1; d < 16; d <<= 1)
            s += __shfl_xor(s, d, 32);
        const int orow = tileM * 16 + v + hi;
        out[(long long)orow * C_DIM + col] = z - m - __logf(s);
    }
}

// ---------------------------------------------------------------------------
// Host-side launcher
// ---------------------------------------------------------------------------
extern "C" void kernel_launch(void* const* d_in, const int* in_sizes, int n_in,
                              void* d_out, int out_size, void* d_ws, size_t ws_size,
                              hipStream_t stream) {
    const float* x   = (const float*)d_in[0];
    const int*   ei  = (const int*)d_in[1];       // [2, E] int32
    const float* W1l = (const float*)d_in[2];
    const float* W1r = (const float*)d_in[3];
    const float* b1  = (const float*)d_in[4];
    const float* W2l = (const float*)d_in[5];
    const float* W2r = (const float*)d_in[6];
    const float* b2  = (const float*)d_in[7];
    const int E = in_sizes[1] / 2;
    const int* src = ei;
    const int* dst = ei + E;

    // workspace layout (floats): [deg | agg1 | agg2 | h]
    float* ws   = (float*)d_ws;
    float* deg  = ws;
    float* agg1 = deg  + N_NODES;
    float* agg2 = agg1 + (size_t)N_NODES * F_IN;
    float* hbuf = agg2 + (size_t)N_NODES * H_DIM;

    // 1) zero deg+agg1+agg2 (contiguous: N*(1+128+256) floats, /4 for float4)
    {
        int n4 = (N_NODES * (1 + F_IN + H_DIM)) / 4;
        zero_ws_kernel<<<(n4 + 255) / 256, 256, 0, stream>>>((float4*)ws, n4);
    }
    // 2) layer-1 aggregation
    {
        int work = E * 32;
        scatter_f128_kernel<<<(work + 255) / 256, 256, 0, stream>>>(
            x, src, dst, agg1, deg, E);
    }
    // 3) layer-1 GEMM + bias + relu
    //    grid = 2 N-groups * ceil(3125/8) M-blocks, 8 waves (M-tiles) / block
    {
        int mblocks = (M_TILES + 7) / 8;             // 391
        sage_gemm1_kernel<<<2 * mblocks, 256, 0, stream>>>(
            x, agg1, deg, W1l, W1r, b1, hbuf);
    }
    // 4) layer-2 aggregation
    {
        int work = E * 64;
        scatter_f256_kernel<<<(work + 255) / 256, 256, 0, stream>>>(
            hbuf, src, dst, agg2, E);
    }
    // 5) layer-2 GEMM + bias + log_softmax
    {
        sage_gemm2_softmax_kernel<<<(M_TILES + 7) / 8, 256, 0, stream>>>(
            hbuf, agg2, deg, W2l, W2r, b2, (float*)d_out);
    }
}